// E3nnVBnet_25744033973176
// MI455X (gfx1250) — compile-verified
//
#include <hip/hip_runtime.h>
#include <hip/hip_bf16.h>
#include <math.h>

#define N_NODES 50000
#define E_EDGES 800000
#define NLAYERS 4
#define NGROUPS 16

typedef __attribute__((ext_vector_type(2))) float v2f;
typedef __attribute__((ext_vector_type(8))) float v8f;

__device__ __forceinline__ float silu_f(float x) { return x / (1.0f + __expf(-x)); }

// ---- WMMA fragment helpers (V_WMMA_F32_16X16X4_F32, wave32) -----------------
// A: 16x4 (MxK). lanes 0-15: M=lane, K={0,1}; lanes 16-31: M=lane-16, K={2,3}.
__device__ __forceinline__ v2f fragA(const float* base, int ldk, int kk, int lane) {
    int m  = lane & 15;
    int k0 = (kk << 2) + ((lane >> 4) << 1);
    const float* p = base + m * ldk + k0;
    v2f r; r.x = p[0]; r.y = p[1];
    return r;
}
// B: 4x16 (KxN), row-major W[k][n]. VGPR0: K={0|2}, VGPR1: K={1|3}; N = lane&15.
__device__ __forceinline__ v2f fragB(const float* W, int ldn, int kk, int nn, int lane) {
    int n  = (nn << 4) + (lane & 15);
    int k0 = (kk << 2) + ((lane >> 4) << 1);
    v2f r; r.x = W[k0 * ldn + n]; r.y = W[(k0 + 1) * ldn + n];
    return r;
}
#define WMMA_F32(a, b, c) \
    __builtin_amdgcn_wmma_f32_16x16x4_f32(false, (a), false, (b), (short)0, (c), false, false)

// ---- zero-init ---------------------------------------------------------------
__global__ void k_zero(float* __restrict__ p, int n) {
    int i = blockIdx.x * blockDim.x + threadIdx.x;
    if (i < n) p[i] = 0.0f;
}

// ---- edge preprocessing: sh1, degree/na1 atomics, 4x gate MLPs ---------------
__global__ __launch_bounds__(256)
void k_edge_pre(const float* __restrict__ edge_attr, const int* __restrict__ dst,
                const float* __restrict__ mlp_w1, const float* __restrict__ mlp_b1,
                const float* __restrict__ mlp_w2, const float* __restrict__ mlp_b2,
                float* __restrict__ sh1, float* __restrict__ gates,
                float* __restrict__ raw, float* __restrict__ na1acc) {
    __shared__ float lw1[NLAYERS * 8 * 64];
    __shared__ float lb1[NLAYERS * 64];
    __shared__ float lw2[NLAYERS * 64];
    __shared__ float lb2[NLAYERS];
    for (int i = threadIdx.x; i < NLAYERS * 8 * 64; i += blockDim.x) lw1[i] = mlp_w1[i];
    for (int i = threadIdx.x; i < NLAYERS * 64; i += blockDim.x) lb1[i] = mlp_b1[i];
    for (int i = threadIdx.x; i < NLAYERS * 64; i += blockDim.x) lw2[i] = mlp_w2[i];
    for (int i = threadIdx.x; i < NLAYERS; i += blockDim.x) lb2[i] = mlp_b2[i];
    __syncthreads();

    int e = blockIdx.x * blockDim.x + threadIdx.x;
    if (e >= E_EDGES) return;

    float ea[10];
#pragma unroll
    for (int i = 0; i < 10; ++i) ea[i] = edge_attr[(size_t)e * 10 + i];
    float rx = ea[7], ry = ea[8], rz = ea[9];
    float rn = sqrtf(rx * rx + ry * ry + rz * rz);
    float rden = fmaxf(rn, 1e-12f);
    const float SQRT3 = 1.7320508075688772f;
    float s0 = SQRT3 * rx / rden, s1 = SQRT3 * ry / rden, s2 = SQRT3 * rz / rden;
    sh1[(size_t)e * 3 + 0] = s0;
    sh1[(size_t)e * 3 + 1] = s1;
    sh1[(size_t)e * 3 + 2] = s2;

    int d = dst[e];
    atomicAdd(&raw[d], 1.0f);
    atomicAdd(&na1acc[d * 3 + 0], s0);
    atomicAdd(&na1acc[d * 3 + 1], s1);
    atomicAdd(&na1acc[d * 3 + 2], s2);

    float es[8] = {ea[0], ea[1], ea[2], ea[3], ea[4], ea[5], ea[6], rn};
#pragma unroll
    for (int l = 0; l < NLAYERS; ++l) {
        float gsum = lb2[l];
        for (int j = 0; j < 64; ++j) {
            float hs = lb1[l * 64 + j];
#pragma unroll
            for (int i = 0; i < 8; ++i) hs += es[i] * lw1[l * 512 + i * 64 + j];
            gsum += silu_f(hs) * lw2[l * 64 + j];
        }
        gates[(size_t)l * E_EDGES + e] = gsum;
    }
}

// ---- node PE + embedding -----------------------------------------------------
__global__ __launch_bounds__(128)
void k_node_embed(const float* __restrict__ x, const float* __restrict__ evecs,
                  const float* __restrict__ evals,
                  const float* __restrict__ phi_w1, const float* __restrict__ phi_b1,
                  const float* __restrict__ phi_w2, const float* __restrict__ phi_b2,
                  const float* __restrict__ rho_w1, const float* __restrict__ rho_b1,
                  const float* __restrict__ rho_w2, const float* __restrict__ rho_b2,
                  const float* __restrict__ emb_w0, const float* __restrict__ emb_w1,
                  const float* __restrict__ raw, const float* __restrict__ na1acc,
                  float* __restrict__ sA, float* __restrict__ vA) {
    __shared__ float lp1[2 * 64];
    __shared__ float lpb1[64];
    __shared__ float lp2[64 * 16];
    __shared__ float lpb2[16];
    __shared__ float lr1[128 * 64];
    __shared__ float lrb1[64];
    __shared__ float lr2[64 * 16];
    __shared__ float lrb2[16];
    __shared__ float le0[64 * 64];
    for (int i = threadIdx.x; i < 128; i += blockDim.x) lp1[i] = phi_w1[i];
    for (int i = threadIdx.x; i < 64; i += blockDim.x) lpb1[i] = phi_b1[i];
    for (int i = threadIdx.x; i < 1024; i += blockDim.x) lp2[i] = phi_w2[i];
    for (int i = threadIdx.x; i < 16; i += blockDim.x) lpb2[i] = phi_b2[i];
    for (int i = threadIdx.x; i < 8192; i += blockDim.x) lr1[i] = rho_w1[i];
    for (int i = threadIdx.x; i < 64; i += blockDim.x) lrb1[i] = rho_b1[i];
    for (int i = threadIdx.x; i < 1024; i += blockDim.x) lr2[i] = rho_w2[i];
    for (int i = threadIdx.x; i < 16; i += blockDim.x) lrb2[i] = rho_b2[i];
    for (int i = threadIdx.x; i < 4096; i += blockDim.x) le0[i] = emb_w0[i];
    __syncthreads();

    int n = blockIdx.x * blockDim.x + threadIdx.x;
    if (n >= N_NODES) return;

    // phi (with sign symmetrization), per eigenvector
    float z[128];
    for (int k = 0; k < 8; ++k) {
        float ev = evecs[(size_t)n * 8 + k];
        float lam = evals[k];
        float acc[16];
#pragma unroll
        for (int o = 0; o < 16; ++o) acc[o] = 2.0f * lpb2[o];
        for (int sgn = 0; sgn < 2; ++sgn) {
            float e0 = sgn ? -ev : ev;
            for (int j = 0; j < 64; ++j) {
                float hh = silu_f(e0 * lp1[j] + lam * lp1[64 + j] + lpb1[j]);
#pragma unroll
                for (int o = 0; o < 16; ++o) acc[o] += hh * lp2[j * 16 + o];
            }
        }
#pragma unroll
        for (int o = 0; o < 16; ++o) z[k * 16 + o] = acc[o];
    }
    // rho
    float V[16];
    {
        float hr[64];
        for (int j = 0; j < 64; ++j) {
            float s = lrb1[j];
            for (int i = 0; i < 128; ++i) s += z[i] * lr1[i * 64 + j];
            hr[j] = silu_f(s);
        }
        for (int o = 0; o < 16; ++o) {
            float s = lrb2[o];
            for (int j = 0; j < 64; ++j) s += hr[j] * lr2[j * 16 + o];
            V[o] = s;
        }
    }
    float xin[64];
    for (int i = 0; i < 48; ++i) xin[i] = x[(size_t)n * 48 + i];
#pragma unroll
    for (int o = 0; o < 16; ++o) xin[48 + o] = V[o];

    float rawn = raw[n];
    float cnt = fmaxf(rawn, 1.0f);
    float na0 = rawn > 0.0f ? 1.0f : 0.0f;
    float nx = na1acc[n * 3 + 0] / cnt;
    float ny = na1acc[n * 3 + 1] / cnt;
    float nz = na1acc[n * 3 + 2] / cnt;

    for (int j = 0; j < 64; ++j) {
        float s = 0.0f;
        for (int u = 0; u < 64; ++u) s += xin[u] * le0[u * 64 + j];
        sA[(size_t)n * 64 + j] = s * na0 * 0.125f;
    }
    for (int vv = 0; vv < 32; ++vv) {
        float s = 0.0f;
        for (int u = 0; u < 64; ++u) s += xin[u] * emb_w1[u * 32 + vv];
        s *= 0.125f;
        vA[(size_t)n * 96 + vv * 3 + 0] = s * nx;
        vA[(size_t)n * 96 + vv * 3 + 1] = s * ny;
        vA[(size_t)n * 96 + vv * 3 + 2] = s * nz;
    }
}

// ---- per-layer edge message kernel: 16-edge tiles via f32 WMMA --------------
#define SJ_LD 68   /* 64 + 4 pad: bank-conflict-free A reads */
#define VD_LD 36   /* 32 + 4 pad */

__global__ __launch_bounds__(64)
void k_edge_message(const int* __restrict__ src, const int* __restrict__ dst,
                    const float* __restrict__ sh1, const float* __restrict__ gates,
                    const float* __restrict__ Wss, const float* __restrict__ Wvv,
                    const float* __restrict__ Wsv, const float* __restrict__ Wvs,
                    const float* __restrict__ sIn, const float* __restrict__ vIn,
                    float* __restrict__ sOut, float* __restrict__ vOut) {
    __shared__ float lWss[64 * 64];
    __shared__ float lWvv[32 * 64];
    __shared__ float lWsv[64 * 32];
    __shared__ float lWvs[32 * 32];
    __shared__ float lSj[2][16 * SJ_LD];
    __shared__ float lDot[2][16 * VD_LD];
    __shared__ float lVj[2][3][16 * VD_LD];
    __shared__ float lSh[2][16 * 4];
    __shared__ float lGate[2][16];
    __shared__ int lDst[2][16];

    const int tid = threadIdx.x;
    for (int i = tid; i < 64 * 64; i += 64) lWss[i] = Wss[i];
    for (int i = tid; i < 32 * 64; i += 64) lWvv[i] = Wvv[i];
    for (int i = tid; i < 64 * 32; i += 64) lWsv[i] = Wsv[i];
    for (int i = tid; i < 32 * 32; i += 64) lWvs[i] = Wvs[i];
    __syncthreads();

    const int w = tid >> 5;
    const int lane = tid & 31;
    const int t = lane >> 1;   // tile-local edge handled by this lane pair
    const int h = lane & 1;    // which half of the feature vector
    const int col = lane & 15;
    const int mofs = (lane >> 4) << 3;

    const int numTiles = (E_EDGES + 15) / 16;
    const int numGroups = (numTiles + 1) / 2;
    const float inv = 0.10206207261596575f;  // 1/sqrt(96)
    const float isq3 = 0.5773502691896258f;  // 1/sqrt(3)

    for (int g = blockIdx.x; g < numGroups; g += gridDim.x) {
        int tile = g * 2 + w;
        int e = tile * 16 + t;
        bool valid = (tile < numTiles) && (e < E_EDGES);

        // ---------- gather into LDS ----------
        int sN = valid ? src[e] : 0;
        if (h == 0) {
            if (valid) {
                lDst[w][t] = dst[e];
                lGate[w][t] = gates[e];
                lSh[w][t * 4 + 0] = sh1[(size_t)e * 3 + 0];
                lSh[w][t * 4 + 1] = sh1[(size_t)e * 3 + 1];
                lSh[w][t * 4 + 2] = sh1[(size_t)e * 3 + 2];
            } else {
                lDst[w][t] = 0;
                lGate[w][t] = 0.0f;
                lSh[w][t * 4 + 0] = 0.0f;
                lSh[w][t * 4 + 1] = 0.0f;
                lSh[w][t * 4 + 2] = 0.0f;
            }
        }
        if (valid) {
            const float4* sp = (const float4*)(sIn + (size_t)sN * 64) + h * 8;
#pragma unroll
            for (int i = 0; i < 8; ++i) {
                float4 q = sp[i];
                float* d = &lSj[w][t * SJ_LD + h * 32 + i * 4];
                d[0] = q.x; d[1] = q.y; d[2] = q.z; d[3] = q.w;
            }
            const float4* vp = (const float4*)(vIn + (size_t)sN * 96) + h * 12;
#pragma unroll
            for (int i = 0; i < 12; ++i) {
                float4 q = vp[i];
                int p = (h * 12 + i) * 4;
                lVj[w][(p + 0) % 3][t * VD_LD + (p + 0) / 3] = q.x;
                lVj[w][(p + 1) % 3][t * VD_LD + (p + 1) / 3] = q.y;
                lVj[w][(p + 2) % 3][t * VD_LD + (p + 2) / 3] = q.z;
                lVj[w][(p + 3) % 3][t * VD_LD + (p + 3) / 3] = q.w;
            }
        } else {
#pragma unroll
            for (int i = 0; i < 32; ++i) lSj[w][t * SJ_LD + h * 32 + i] = 0.0f;
#pragma unroll
            for (int i = 0; i < 48; ++i) {
                int p = h * 48 + i;
                lVj[w][p % 3][t * VD_LD + p / 3] = 0.0f;
            }
        }
        __syncthreads();

        // dot[e][u] = sum_m vj[u][m]*sh1[m] / sqrt(3)
        {
            float sx = lSh[w][t * 4 + 0], sy = lSh[w][t * 4 + 1], sz = lSh[w][t * 4 + 2];
#pragma unroll
            for (int i = 0; i < 16; ++i) {
                int u = h * 16 + i;
                float d = lVj[w][0][t * VD_LD + u] * sx + lVj[w][1][t * VD_LD + u] * sy +
                          lVj[w][2][t * VD_LD + u] * sz;
                lDot[w][t * VD_LD + u] = d * isq3;
            }
        }
        __syncthreads();

        // per-lane per-row gate/dst (D layout: M = r + 8*(lane>=16))
        int dN8[8];
        float g8[8];
#pragma unroll
        for (int r = 0; r < 8; ++r) {
            int tt = r + mofs;
            dN8[r] = lDst[w][tt];
            g8[r] = lGate[w][tt] * inv;
        }

        const float* mySj = &lSj[w][0];
        const float* myDot = &lDot[w][0];

        // ---------- m_s = gate*inv*(sj@Wss + dot@Wvv) ----------
#pragma unroll
        for (int nn = 0; nn < 4; ++nn) {
            v8f acc = {0, 0, 0, 0, 0, 0, 0, 0};
#pragma unroll
            for (int kk = 0; kk < 16; ++kk) {
                v2f a = fragA(mySj, SJ_LD, kk, lane);
                v2f b = fragB(lWss, 64, kk, nn, lane);
                acc = WMMA_F32(a, b, acc);
            }
#pragma unroll
            for (int kk = 0; kk < 8; ++kk) {
                v2f a = fragA(myDot, VD_LD, kk, lane);
                v2f b = fragB(lWvv, 64, kk, nn, lane);
                acc = WMMA_F32(a, b, acc);
            }
#pragma unroll
            for (int r = 0; r < 8; ++r)
                atomicAdd(&sOut[(size_t)dN8[r] * 64 + nn * 16 + col], g8[r] * acc[r]);
        }

        // ---------- T_sv = sj@Wsv (kept in regs) ----------
        v8f accSV[2];
#pragma unroll
        for (int nn = 0; nn < 2; ++nn) {
            v8f acc = {0, 0, 0, 0, 0, 0, 0, 0};
#pragma unroll
            for (int kk = 0; kk < 16; ++kk) {
                v2f a = fragA(mySj, SJ_LD, kk, lane);
                v2f b = fragB(lWsv, 32, kk, nn, lane);
                acc = WMMA_F32(a, b, acc);
            }
            accSV[nn] = acc;
        }

        // ---------- m_v[:, :, m] = gate*inv*(T_sv*sh1_m + vj_m@Wvs) ----------
#pragma unroll
        for (int m = 0; m < 3; ++m) {
            const float* myV = &lVj[w][m][0];
            float sh8[8];
#pragma unroll
            for (int r = 0; r < 8; ++r) sh8[r] = lSh[w][(r + mofs) * 4 + m];
#pragma unroll
            for (int nn = 0; nn < 2; ++nn) {
                v8f acc = {0, 0, 0, 0, 0, 0, 0, 0};
#pragma unroll
                for (int kk = 0; kk < 8; ++kk) {
                    v2f a = fragA(myV, VD_LD, kk, lane);
                    v2f b = fragB(lWvs, 32, kk, nn, lane);
                    acc = WMMA_F32(a, b, acc);
                }
#pragma unroll
                for (int r = 0; r < 8; ++r) {
                    float val = g8[r] * (accSV[nn][r] * sh8[r] + acc[r]);
                    atomicAdd(&vOut[(size_t)dN8[r] * 96 + (nn * 16 + col) * 3 + m], val);
                }
            }
        }
        __syncthreads();  // protect staging LDS before next group iteration
    }
}

// ---- readout -----------------------------------------------------------------
__global__ __launch_bounds__(256)
void k_readout(const float* __restrict__ sF, const float* __restrict__ out_w,
               const int* __restrict__ batch, float* __restrict__ gsum,
               float* __restrict__ gcnt) {
    int n = blockIdx.x * blockDim.x + threadIdx.x;
    if (n >= N_NODES) return;
    float o = 0.0f;
    for (int j = 0; j < 64; ++j) o += sF[(size_t)n * 64 + j] * out_w[j];
    o *= 0.125f;
    int b = batch[n];
    atomicAdd(&gsum[b], o);
    atomicAdd(&gcnt[b], 1.0f);
}

__global__ void k_final(const float* __restrict__ gsum, const float* __restrict__ gcnt,
                        float* __restrict__ out) {
    int g = threadIdx.x;
    if (g < NGROUPS) out[g] = gsum[g] / fmaxf(gcnt[g], 1.0f);
}

// ---- host launcher -----------------------------------------------------------
extern "C" void kernel_launch(void* const* d_in, const int* in_sizes, int n_in,
                              void* d_out, int out_size, void* d_ws, size_t ws_size,
                              hipStream_t stream) {
    const float* x = (const float*)d_in[0];
    const float* edge_attr = (const float*)d_in[1];
    const float* lap_evecs = (const float*)d_in[2];
    const float* lap_evals = (const float*)d_in[3];
    const float* phi_w1 = (const float*)d_in[4];
    const float* phi_b1 = (const float*)d_in[5];
    const float* phi_w2 = (const float*)d_in[6];
    const float* phi_b2 = (const float*)d_in[7];
    const float* rho_w1 = (const float*)d_in[8];
    const float* rho_b1 = (const float*)d_in[9];
    const float* rho_w2 = (const float*)d_in[10];
    const float* rho_b2 = (const float*)d_in[11];
    const float* emb_w0 = (const float*)d_in[12];
    const float* emb_w1 = (const float*)d_in[13];
    const float* tp_w_ss = (const float*)d_in[14];
    const float* tp_w_vv = (const float*)d_in[15];
    const float* tp_w_sv = (const float*)d_in[16];
    const float* tp_w_vs = (const float*)d_in[17];
    const float* mlp_w1 = (const float*)d_in[18];
    const float* mlp_b1 = (const float*)d_in[19];
    const float* mlp_w2 = (const float*)d_in[20];
    const float* mlp_b2 = (const float*)d_in[21];
    const float* out_w = (const float*)d_in[22];
    const int* edge_index = (const int*)d_in[23];
    const int* batch = (const int*)d_in[24];
    const int* src = edge_index;
    const int* dst = edge_index + E_EDGES;

    // workspace layout (floats)
    float* ws = (float*)d_ws;
    size_t off = 0;
    float* raw = ws + off;    off += N_NODES;
    float* na1acc = ws + off; off += 3 * (size_t)N_NODES;
    float* gsum = ws + off;   off += NGROUPS;
    float* gcnt = ws + off;   off += NGROUPS;
    float* sh1 = ws + off;    off += 3 * (size_t)E_EDGES;
    float* gates = ws + off;  off += (size_t)NLAYERS * E_EDGES;
    float* sA = ws + off;     off += 64 * (size_t)N_NODES;
    float* sB = ws + off;     off += 64 * (size_t)N_NODES;
    float* vA = ws + off;     off += 96 * (size_t)N_NODES;
    float* vB = ws + off;     off += 96 * (size_t)N_NODES;

    int zn = 4 * N_NODES + 2 * NGROUPS;
    k_zero<<<(zn + 255) / 256, 256, 0, stream>>>(ws, zn);

    k_edge_pre<<<(E_EDGES + 255) / 256, 256, 0, stream>>>(
        edge_attr, dst, mlp_w1, mlp_b1, mlp_w2, mlp_b2, sh1, gates, raw, na1acc);

    k_node_embed<<<(N_NODES + 127) / 128, 128, 0, stream>>>(
        x, lap_evecs, lap_evals, phi_w1, phi_b1, phi_w2, phi_b2, rho_w1, rho_b1,
        rho_w2, rho_b2, emb_w0, emb_w1, raw, na1acc, sA, vA);

    float* sIn = sA; float* vIn = vA;
    float* sOut = sB; float* vOut = vB;
    for (int l = 0; l < NLAYERS; ++l) {
        hipMemcpyAsync(sOut, sIn, 64 * (size_t)N_NODES * sizeof(float),
                       hipMemcpyDeviceToDevice, stream);
        hipMemcpyAsync(vOut, vIn, 96 * (size_t)N_NODES * sizeof(float),
                       hipMemcpyDeviceToDevice, stream);
        k_edge_message<<<2048, 64, 0, stream>>>(
            src, dst, sh1, gates + (size_t)l * E_EDGES,
            tp_w_ss + (size_t)l * 64 * 64, tp_w_vv + (size_t)l * 32 * 64,
            tp_w_sv + (size_t)l * 64 * 32, tp_w_vs + (size_t)l * 32 * 32,
            sIn, vIn, sOut, vOut);
        float* ts = sIn; sIn = sOut; sOut = ts;
        float* tv = vIn; vIn = vOut; vOut = tv;
    }

    k_readout<<<(N_NODES + 255) / 256, 256, 0, stream>>>(sIn, out_w, batch, gsum, gcnt);
    k_final<<<1, 32, 0, stream>>>(gsum, gcnt, (float*)d_out);
}